// Max_margin_loss_40398462386415
// MI455X (gfx1250) — compile-verified
//
#include <hip/hip_runtime.h>
#include <hip/hip_bf16.h>

typedef __attribute__((ext_vector_type(2))) float v2f;
typedef __attribute__((ext_vector_type(8))) float v8f;

#define DIM 4096
#define BN  8192
#define MARGIN 0.5f
#define EPSN 1e-8f

// workspace layout (float offsets)
#define WS_DOT  0
#define WS_NRM  8192
#define WS_N1SQ 16384
#define WS_CNEG 24576
#define WS_CPOS 32768

// ---------------- k0: anchor norm^2 + zero the output ----------------
__global__ __launch_bounds__(256)
void k0_anchor(const float* __restrict__ v1, float* __restrict__ ws,
               float* __restrict__ out) {
    __shared__ float red[256];
    float s = 0.f;
    for (int t = threadIdx.x; t < DIM; t += 256) { float x = v1[t]; s += x * x; }
    red[threadIdx.x] = s;
    __syncthreads();
    for (int off = 128; off > 0; off >>= 1) {
        if (threadIdx.x < off) red[threadIdx.x] += red[threadIdx.x + off];
        __syncthreads();
    }
    if (threadIdx.x == 0) { ws[WS_N1SQ] = red[0]; out[0] = 0.f; }
}

// ---------------- k1: WMMA GEMV + row norms ----------------
// One wave32 handles a tile of 16 rows.
//   A (16x4 f32, documented layout): anchor chunk replicated over M.
//     lane l (h = l>>4): a = {anchor[kk+2h], anchor[kk+2h+1]}
//   B (4x16 f32, layout assumed VGPRv <-> K=v (lanes0-15), K=v+2 (lanes16-31)):
//     column n = row_{base+n}: lane l (n = l&15): b = {row_n[kk+2h], row_n[kk+2h+1]}
//   => C[m][n] = dot(row_{base+n}, anchor) for every m; read C[0][n] (acc[0], lanes 0-15).
// Row-norm partials: lane l covers k with floor((k mod 4)/2)==h -> combine l with l^16.
__global__ __launch_bounds__(128)
void k1_gemv(const float* __restrict__ v1, const float* __restrict__ v2,
             float* __restrict__ ws) {
    const int lane = threadIdx.x & 31;
    const int wave = threadIdx.x >> 5;
    const int tile = blockIdx.x * 4 + wave;        // 512 tiles of 16 rows
    const int n    = lane & 15;
    const int h    = lane >> 4;

    const float* __restrict__ rowp = v2 + (size_t)(tile * 16 + n) * DIM;

    v8f   acc = {};
    float nrm = 0.f;

#pragma unroll 8
    for (int kk = 0; kk < DIM; kk += 4) {
        v2f a = *(const v2f*)(v1   + kk + 2 * h);
        v2f b = *(const v2f*)(rowp + kk + 2 * h);
        nrm += b.x * b.x + b.y * b.y;
        // 8 args: (neg_a, A, neg_b, B, c_mod, C, reuse_a, reuse_b)
        acc = __builtin_amdgcn_wmma_f32_16x16x4_f32(
            false, a, false, b, (short)0, acc, false, false);
    }

    float other = __shfl_xor(nrm, 16, 32);   // combine the two K-half partials
    if (h == 0) {
        ws[WS_DOT + tile * 16 + n] = acc[0];         // C[0][n]
        ws[WS_NRM + tile * 16 + n] = nrm + other;    // ||row||^2
    }
}

// ---------------- k1b: cosine + label folding ----------------
__global__ __launch_bounds__(256)
void k1b_cos(const int* __restrict__ label, float* __restrict__ ws) {
    int b = blockIdx.x * blockDim.x + threadIdx.x;
    if (b >= BN) return;
    float n1 = fmaxf(sqrtf(ws[WS_N1SQ]), EPSN);
    float n2 = fmaxf(sqrtf(ws[WS_NRM + b]), EPSN);
    float c  = ws[WS_DOT + b] / (n1 * n2);
    int lab  = label[b];
    ws[WS_CNEG + b] = (lab == 0) ? c :  -1e30f;  // relu(-1e30 + x) == 0
    ws[WS_CPOS + b] = (lab == 1) ? c :   1e30f;  // relu(x - 1e30) == 0
}

// ---------------- k2: pairwise hinge sum ----------------
__global__ __launch_bounds__(256)
void k2_pair(const float* __restrict__ ws, float* __restrict__ out) {
    __shared__ float cn[BN];       // 32 KB: cneg for all j
    __shared__ float red[256];
    for (int t = threadIdx.x; t < BN; t += 256) cn[t] = ws[WS_CNEG + t];
    __syncthreads();

    const int i = blockIdx.x * 256 + threadIdx.x;   // 32 blocks * 256 = 8192
    const float ci = ws[WS_CPOS + i];
    float s = 0.f;
#pragma unroll 8
    for (int j = 0; j < BN; ++j)
        s += fmaxf(cn[j] - ci + MARGIN, 0.f);

    red[threadIdx.x] = s;
    __syncthreads();
    for (int off = 128; off > 0; off >>= 1) {
        if (threadIdx.x < off) red[threadIdx.x] += red[threadIdx.x + off];
        __syncthreads();
    }
    if (threadIdx.x == 0) atomicAdd(out, red[0]);
}

extern "C" void kernel_launch(void* const* d_in, const int* in_sizes, int n_in,
                              void* d_out, int out_size, void* d_ws, size_t ws_size,
                              hipStream_t stream) {
    const float* v1    = (const float*)d_in[0];   // [4096]
    const float* v2    = (const float*)d_in[1];   // [8192, 4096]
    const int*   label = (const int*)d_in[2];     // [8192]
    float* out = (float*)d_out;                   // [1]
    float* ws  = (float*)d_ws;

    k0_anchor<<<1,   256, 0, stream>>>(v1, ws, out);
    k1_gemv  <<<128, 128, 0, stream>>>(v1, v2, ws);
    k1b_cos  <<<32,  256, 0, stream>>>(label, ws);
    k2_pair  <<<32,  256, 0, stream>>>(ws, out);
}